// ViewRenderer_2267742732617
// MI455X (gfx1250) — compile-verified
//
#include <hip/hip_runtime.h>
#include <math.h>

// ---------------- problem constants (from reference setup_inputs) -------------
#define B_  2
#define N_  6
#define H_  384
#define W_  640
#define HW_ (H_*W_)
#define EPS_Z 1e-7f
#define EPS_M 1e-8f
#define EPS_V 1e-16f

typedef float v2f __attribute__((ext_vector_type(2)));
typedef float v8f __attribute__((ext_vector_type(8)));

// ---------------- workspace layout (floats) -----------------------------------
// [0,192)    : 12 C2 matrices (pair p = w*2+b), 16 floats each, C2 = K*T*invK4
// [192,252)  : per-pair sums A1..A5 (5 each)
// [252,256)  : per-batch ref sums R1,R2
// [256,316)  : per-pair stats s_mean,s_std,w_mean,w_std,any_zero
#define WS_MAT   0
#define WS_SUMS  192
#define WS_REF   252
#define WS_STATS 256

#if __has_builtin(__builtin_amdgcn_wmma_f32_16x16x4_f32)
#define HAVE_WMMA44 1
#else
#define HAVE_WMMA44 0
#endif

__device__ __forceinline__ int imin(int a, int b) { return a < b ? a : b; }
__device__ __forceinline__ int imax(int a, int b) { return a > b ? a : b; }

// warp index -> source camera: w<2 -> ci ; w==2||w==4 -> nb0 ; else nb1
__device__ __forceinline__ int warp_cam(int w, int ci, int nb0, int nb1) {
    return (w < 2) ? ci : ((w == 2 || w == 4) ? nb0 : nb1);
}

// ---------------- setup: C2 = K * T * invK4 for all 12 (warp,batch) pairs -----
// One wave (32 threads). WMMA path: A = four stacked 4x4 K blocks (16x4),
// B = four side-by-side 4x4 T blocks (4x16); diagonal 4x4 blocks of D are K_i*T_i.
// Second WMMA multiplies those diag blocks (restriped via LDS) by invK4 blocks.
__global__ void setup_mats(const float* __restrict__ K,
                           const float* __restrict__ invK,
                           const float* __restrict__ p2c,
                           const float* __restrict__ c2n,
                           const float* __restrict__ rel,
                           const int*   __restrict__ cam_index,
                           const int*   __restrict__ nbrs,
                           float* __restrict__ ws)
{
    const int lane = threadIdx.x;            // 0..31, one wave, EXEC all ones
    const int ci  = cam_index[0];
    const int nb0 = nbrs[0], nb1 = nbrs[1];

#if HAVE_WMMA44
    __shared__ float lds[256];               // 16x16 staging of D between WMMAs

    const int m    = lane & 15;              // A row / D column selector
    const int half = lane >> 4;              // 0: K=0,1  1: K=2,3 (16x4 f32 A layout)
    const int k0   = half * 2;
    const int i    = m >> 2;                 // pair-in-group (A side)
    const int r    = m & 3;                  // row within 4x4 block
    const int j    = m >> 2;                 // pair-in-group (B side, col n=m)
    const int c    = m & 3;                  // col within 4x4 block

    for (int g = 0; g < 3; ++g) {
        // ---- A: stacked K blocks, A[4i+r][k] = K_pair[r][k]
        int pairA = g * 4 + i;
        int bA = pairA & 1, wA = pairA >> 1;
        int camA = warp_cam(wA, ci, nb0, nb1);
        const float* Kp = K + (size_t)(bA * N_ + camA) * 16;
        v2f a; a.x = Kp[r * 4 + k0]; a.y = Kp[r * 4 + k0 + 1];

        // ---- B: side-by-side T blocks, B[k][4j+c] = T_pair[k][c]
        int pairB = g * 4 + j;
        int bB = pairB & 1, wB = pairB >> 1;
        const float* Tp;
        if (wB == 0)      Tp = p2c + (size_t)(bB * N_ + ci) * 16;
        else if (wB == 1) Tp = c2n + (size_t)(bB * N_ + ci) * 16;
        else {
            int f  = (wB >= 4) ? 1 : 0;
            int nn = (wB == 2 || wB == 4) ? 0 : 1;
            Tp = rel + (size_t)((f * 2 + nn) * B_ + bB) * 16;
        }
        v2f bm; bm.x = Tp[k0 * 4 + c]; bm.y = Tp[(k0 + 1) * 4 + c];

        v8f acc = {};
        acc = __builtin_amdgcn_wmma_f32_16x16x4_f32(false, a, false, bm,
                                                    (short)0, acc, false, false);
        // scatter D (C1 = block products) to LDS: VGPR e holds D[e+half*8][m]
        #pragma unroll
        for (int e = 0; e < 8; ++e)
            lds[(e + half * 8) * 16 + m] = acc[e];
        __syncthreads();

        // ---- A2[4i+r][k] = C1_diag_i[r][k] = lds[(4i+r)*16 + 4i + k]
        v2f a2; a2.x = lds[m * 16 + (i << 2) + k0];
                a2.y = lds[m * 16 + (i << 2) + k0 + 1];
        // ---- B2[k][4j+c] = invK4 of pair j (invK always ref camera)
        const float* Ip = invK + (size_t)(bB * N_ + ci) * 16;
        v2f b2; b2.x = Ip[k0 * 4 + c]; b2.y = Ip[(k0 + 1) * 4 + c];

        v8f acc2 = {};
        acc2 = __builtin_amdgcn_wmma_f32_16x16x4_f32(false, a2, false, b2,
                                                     (short)0, acc2, false, false);
        // extract diagonal 4x4 blocks -> C2 of each pair
        #pragma unroll
        for (int e = 0; e < 8; ++e) {
            int row = e + half * 8;
            int col = m;
            if ((row >> 2) == (col >> 2)) {
                int q = row >> 2;
                ws[WS_MAT + (g * 4 + q) * 16 + (row & 3) * 4 + (col & 3)] = acc2[e];
            }
        }
        __syncthreads();
    }
#else
    // scalar fallback: thread p computes one pair
    int p = lane;
    if (p < 12) {
        int b = p & 1, w = p >> 1;
        int cam = warp_cam(w, ci, nb0, nb1);
        const float* Kp = K + (size_t)(b * N_ + cam) * 16;
        const float* Ip = invK + (size_t)(b * N_ + ci) * 16;
        const float* Tp;
        if (w == 0)      Tp = p2c + (size_t)(b * N_ + ci) * 16;
        else if (w == 1) Tp = c2n + (size_t)(b * N_ + ci) * 16;
        else {
            int f  = (w >= 4) ? 1 : 0;
            int nn = (w == 2 || w == 4) ? 0 : 1;
            Tp = rel + (size_t)((f * 2 + nn) * B_ + b) * 16;
        }
        float C1[16];
        for (int r = 0; r < 4; ++r)
            for (int cc = 0; cc < 4; ++cc) {
                float s = 0.f;
                for (int k = 0; k < 4; ++k) s += Kp[r * 4 + k] * Tp[k * 4 + cc];
                C1[r * 4 + cc] = s;
            }
        for (int r = 0; r < 4; ++r)
            for (int cc = 0; cc < 4; ++cc) {
                float s = 0.f;
                for (int k = 0; k < 4; ++k) s += C1[r * 4 + k] * Ip[k * 4 + cc];
                ws[WS_MAT + p * 16 + r * 4 + cc] = s;
            }
    }
#endif
}

// ---------------- per-pixel warp: project depth, bilinear img + nearest mask --
__device__ __forceinline__ void warp_one(
    const float* __restrict__ img, const float* __restrict__ maskp,
    int b, int cam, const float* __restrict__ Mt,
    float xf, float yf, float d,
    float& o0, float& o1, float& o2, float& om)
{
    float Z = (Mt[8] * xf + Mt[9] * yf + Mt[10]) * d + Mt[11];
    float inv = 1.0f / (Z + EPS_Z);
    float X = ((Mt[0] * xf + Mt[1] * yf + Mt[2]) * d + Mt[3]) * inv;
    float Y = ((Mt[4] * xf + Mt[5] * yf + Mt[6]) * d + Mt[7]) * inv;
    float pxn = (X * (1.0f / (W_ - 1)) - 0.5f) * 2.0f;
    float pyn = (Y * (1.0f / (H_ - 1)) - 0.5f) * 2.0f;
    bool invalid = (pxn > 1.f) || (pxn < -1.f) || (pyn > 1.f) || (pyn < -1.f);
    float x = (pxn + 1.f) * 0.5f * (W_ - 1);
    float y = (pyn + 1.f) * 0.5f * (H_ - 1);
    float x0 = floorf(x), y0 = floorf(y);
    int xi = (int)x0, yi = (int)y0;
    float wx = x - x0, wy = y - y0;
    float w00 = (1.f - wx) * (1.f - wy), w10 = wx * (1.f - wy);
    float w01 = (1.f - wx) * wy,         w11 = wx * wy;
    int xc0 = imin(imax(xi, 0), W_ - 1), xc1 = imin(imax(xi + 1, 0), W_ - 1);
    int yc0 = imin(imax(yi, 0), H_ - 1), yc1 = imin(imax(yi + 1, 0), H_ - 1);
    float vy0 = (yi >= 0 && yi < H_) ? 1.f : 0.f;
    float vy1 = (yi + 1 >= 0 && yi + 1 < H_) ? 1.f : 0.f;
    float vx0 = (xi >= 0 && xi < W_) ? 1.f : 0.f;
    float vx1 = (xi + 1 >= 0 && xi + 1 < W_) ? 1.f : 0.f;
    float f00 = w00 * vy0 * vx0, f10 = w10 * vy0 * vx1;
    float f01 = w01 * vy1 * vx0, f11 = w11 * vy1 * vx1;
    const float* ib = img + (size_t)(b * N_ + cam) * 3 * HW_;
    size_t i00 = (size_t)yc0 * W_ + xc0, i10 = (size_t)yc0 * W_ + xc1;
    size_t i01 = (size_t)yc1 * W_ + xc0, i11 = (size_t)yc1 * W_ + xc1;
    float o[3];
    #pragma unroll
    for (int c = 0; c < 3; ++c) {
        const float* pc = ib + (size_t)c * HW_;
        float v = pc[i00] * f00 + pc[i10] * f10 + pc[i01] * f01 + pc[i11] * f11;
        o[c] = (v != v) ? 2.0f : v;   // NaN -> 2.0
    }
    o0 = o[0]; o1 = o[1]; o2 = o[2];
    int xn = (int)rintf(x), yn = (int)rintf(y);
    bool mv = (yn >= 0) && (yn < H_) && (xn >= 0) && (xn < W_);
    float mval = maskp[(size_t)(b * N_ + cam) * HW_ +
                       (size_t)imin(imax(yn, 0), H_ - 1) * W_ + imin(imax(xn, 0), W_ - 1)];
    float mw = mv ? mval : 0.f;
    mw = (mw != mw) ? 0.f : mw;       // NaN -> 0
    om = invalid ? 0.f : mw;
}

__device__ __forceinline__ float wave_sum(float v) {
    #pragma unroll
    for (int off = 16; off > 0; off >>= 1) v += __shfl_down(v, off, 32);
    return v;
}

// ---------------- init: zero reduction scratch --------------------------------
__global__ void init_ws(float* __restrict__ ws) {
    int i = threadIdx.x;
    if (i < 64) ws[WS_SUMS + i] = 0.f;
}

// ---------------- pass 1: masked/unmasked reductions per (warp,batch) ---------
// Reduction hierarchy: wave shuffle -> LDS ds_add_f32 (per block, 32 slots)
// -> one global fp32 atomic per block per quantity (8x fewer global atomics).
__global__ void __launch_bounds__(256)
sum_kernel(const float* __restrict__ prev, const float* __restrict__ cur,
           const float* __restrict__ next, const float* __restrict__ maskp,
           const float* __restrict__ depth, const int* __restrict__ cam_index,
           const int* __restrict__ nbrs, float* __restrict__ ws)
{
    __shared__ float sacc[32];               // q = w*5+j (0..29), 30=R1, 31=R2
    if (threadIdx.x < 32) sacc[threadIdx.x] = 0.f;
    __syncthreads();

    int idx = blockIdx.x * blockDim.x + threadIdx.x;     // exact grid: B*HW
    int b = idx / HW_;                                   // uniform per block
    int pix = idx - b * HW_;
    int ci = cam_index[0];
    int nb0 = nbrs[0], nb1 = nbrs[1];
    float xf = (float)(pix % W_), yf = (float)(pix / W_);

    const float* dp = depth + (size_t)(b * N_ + ci) * HW_;
    const float* mp = maskp + (size_t)(b * N_ + ci) * HW_;
    const float* refp = cur + (size_t)(b * N_ + ci) * 3 * HW_;
    // prefetch next block's streaming lines (global_prefetch_b8)
    __builtin_prefetch(dp + pix + 256, 0, 1);
    __builtin_prefetch(mp + pix + 256, 0, 1);
    __builtin_prefetch(refp + pix + 256, 0, 1);
    __builtin_prefetch(refp + HW_ + pix + 256, 0, 1);
    __builtin_prefetch(refp + 2 * HW_ + pix + 256, 0, 1);

    float d  = dp[pix];
    float rm = mp[pix];
    float r0 = refp[pix], r1 = refp[HW_ + pix], r2 = refp[2 * HW_ + pix];
    float rs = r0 + r1 + r2;
    float rs2 = r0 * r0 + r1 * r1 + r2 * r2;
    int lane = threadIdx.x & 31;

    #pragma unroll
    for (int w = 0; w < 6; ++w) {
        const float* img = (w == 1 || w >= 4) ? next : prev;
        int cam = warp_cam(w, ci, nb0, nb1);
        int p = w * 2 + b;
        const float* Mt = ws + WS_MAT + p * 16;
        float o0, o1, o2, om;
        warp_one(img, maskp, b, cam, Mt, xf, yf, d, o0, o1, o2, om);
        float mf = (rm * om > 0.f) ? 1.f : 0.f;
        float wsum = o0 + o1 + o2;
        float q0 = wave_sum(mf);
        float q1 = wave_sum(mf * rs);
        float q2 = wave_sum(mf * wsum);
        float q3 = wave_sum(wsum);
        float q4 = wave_sum(o0 * o0 + o1 * o1 + o2 * o2);
        if (lane == 0) {                      // LDS float atomics: ds_add_f32
            atomicAdd(&sacc[w * 5 + 0], q0);
            atomicAdd(&sacc[w * 5 + 1], q1);
            atomicAdd(&sacc[w * 5 + 2], q2);
            atomicAdd(&sacc[w * 5 + 3], q3);
            atomicAdd(&sacc[w * 5 + 4], q4);
        }
    }
    float t1 = wave_sum(rs), t2 = wave_sum(rs2);
    if (lane == 0) {
        atomicAdd(&sacc[30], t1);
        atomicAdd(&sacc[31], t2);
    }
    __syncthreads();

    // one global atomic per block per quantity
    int q = threadIdx.x;
    if (q < 30) {
        int w = q / 5, j = q - w * 5;
        unsafeAtomicAdd(ws + WS_SUMS + (w * 2 + b) * 5 + j, sacc[q]);
    } else if (q == 30) {
        unsafeAtomicAdd(ws + WS_REF + b * 2 + 0, sacc[30]);
    } else if (q == 31) {
        unsafeAtomicAdd(ws + WS_REF + b * 2 + 1, sacc[31]);
    }
}

// ---------------- pass 2: means / stds / any_zero -----------------------------
__global__ void stats_kernel(float* __restrict__ ws) {
    int p = threadIdx.x;
    if (p >= 12) return;
    int b = p & 1, w = p >> 1;
    const float* S = ws + WS_SUMS + p * 5;
    float A1 = S[0], A2 = S[1], A3 = S[2], A4 = S[3], A5 = S[4];
    float R1 = ws[WS_REF + b * 2], R2 = ws[WS_REF + b * 2 + 1];
    float n = 3.0f * (float)HW_;
    float msum = 3.0f * A1;
    float s_mean = A2 / (msum + EPS_M);
    float w_mean = A3 / (msum + EPS_M);
    float s_var = (R2 - 2.f * s_mean * R1 + n * s_mean * s_mean) / n;
    float w_var = (A5 - 2.f * w_mean * A4 + n * w_mean * w_mean) / n;
    float s_std = sqrtf(s_var + EPS_V);
    float w_std = sqrtf(w_var + EPS_V);
    float A1o = ws[WS_SUMS + (w * 2 + (1 - b)) * 5];
    float anyz = ((3.f * A1 == 0.f) || (3.f * A1o == 0.f)) ? 1.f : 0.f;
    float* O = ws + WS_STATS + p * 5;
    O[0] = s_mean; O[1] = s_std; O[2] = w_mean; O[3] = w_std; O[4] = anyz;
}

// ---------------- pass 3: recompute warps, normalize, write 16 channels -------
__global__ void __launch_bounds__(256)
render_kernel(const float* __restrict__ prev, const float* __restrict__ cur,
              const float* __restrict__ next, const float* __restrict__ maskp,
              const float* __restrict__ depth, const int* __restrict__ cam_index,
              const int* __restrict__ nbrs, const float* __restrict__ ws,
              float* __restrict__ out)
{
    int idx = blockIdx.x * blockDim.x + threadIdx.x;
    int b = idx / HW_;
    int pix = idx - b * HW_;
    int ci = cam_index[0];
    int nb0 = nbrs[0], nb1 = nbrs[1];
    float xf = (float)(pix % W_), yf = (float)(pix / W_);

    const float* dp = depth + (size_t)(b * N_ + ci) * HW_;
    __builtin_prefetch(dp + pix + 256, 0, 1);
    float d = dp[pix];

    float ch[16];
    #pragma unroll
    for (int c = 8; c < 16; ++c) ch[c] = 0.f;

    #pragma unroll
    for (int w = 0; w < 6; ++w) {
        const float* img = (w == 1 || w >= 4) ? next : prev;
        int cam = warp_cam(w, ci, nb0, nb1);
        int p = w * 2 + b;
        const float* Mt = ws + WS_MAT + p * 16;
        float o0, o1, o2, om;
        warp_one(img, maskp, b, cam, Mt, xf, yf, d, o0, o1, o2, om);
        const float* st = ws + WS_STATS + p * 5;
        float sm = st[0], ss = st[1], wmn = st[2], wsd = st[3];
        bool anyz = st[4] != 0.f;
        float scale = ss / (wsd + EPS_M);
        float n0 = anyz ? o0 : ((o0 - wmn) * scale + sm) * om;
        float n1 = anyz ? o1 : ((o1 - wmn) * scale + sm) * om;
        float n2 = anyz ? o2 : ((o2 - wmn) * scale + sm) * om;
        if (w == 0)      { ch[0] = n0; ch[1] = n1; ch[2] = n2; ch[3] = om; }
        else if (w == 1) { ch[4] = n0; ch[5] = n1; ch[6] = n2; ch[7] = om; }
        else if (w < 4)  { ch[8] += n0; ch[9] += n1; ch[10] += n2; ch[11] += om; }
        else             { ch[12] += n0; ch[13] += n1; ch[14] += n2; ch[15] += om; }
    }
    float* ob = out + (size_t)b * 16 * HW_ + pix;
    #pragma unroll
    for (int c = 0; c < 16; ++c) ob[(size_t)c * HW_] = ch[c];
}

// ---------------- host launcher ----------------------------------------------
extern "C" void kernel_launch(void* const* d_in, const int* in_sizes, int n_in,
                              void* d_out, int out_size, void* d_ws, size_t ws_size,
                              hipStream_t stream)
{
    const float* prev  = (const float*)d_in[0];
    const float* cur   = (const float*)d_in[1];
    const float* next  = (const float*)d_in[2];
    const float* maskp = (const float*)d_in[3];
    const float* K     = (const float*)d_in[4];
    const float* invK  = (const float*)d_in[5];
    const float* depth = (const float*)d_in[6];
    const float* p2c   = (const float*)d_in[7];
    const float* c2n   = (const float*)d_in[8];
    const float* rel   = (const float*)d_in[9];
    const int*   camix = (const int*)d_in[10];
    const int*   nbrs  = (const int*)d_in[11];
    float* ws  = (float*)d_ws;
    float* out = (float*)d_out;

    const int nthreads = B_ * HW_;           // 491520, divisible by 256
    const int blocks = nthreads / 256;

    init_ws<<<1, 64, 0, stream>>>(ws);
    setup_mats<<<1, 32, 0, stream>>>(K, invK, p2c, c2n, rel, camix, nbrs, ws);
    sum_kernel<<<blocks, 256, 0, stream>>>(prev, cur, next, maskp, depth,
                                           camix, nbrs, ws);
    stats_kernel<<<1, 32, 0, stream>>>(ws);
    render_kernel<<<blocks, 256, 0, stream>>>(prev, cur, next, maskp, depth,
                                              camix, nbrs, ws, out);
    (void)in_sizes; (void)n_in; (void)out_size; (void)ws_size;
}